// DSVFCell_37641093382422
// MI455X (gfx1250) — compile-verified
//
#include <hip/hip_runtime.h>
#include <hip/hip_bf16.h>

// ---------------------------------------------------------------------------
// DSVF (state-variable filter) scan:  y[B,T], v_final[B,2]
// B=2048, T=8192.  Linear recurrence v' = M v + k x  =>  segmented scan:
//   pass1: per (row,seg) particular end-state from v=0        (reads x once)
//   pass2: per row combine across segments with M^L (2x2)     (tiny)
//   pass3: per (row,seg) rerun from exact start, emit y       (reads x, writes y)
// x tiles are staged into LDS by the Tensor Data Mover (tensor_load_to_lds)
// with pad_enable -> row stride 33 dwords (bank-conflict free), double buffered
// and synchronized with s_wait_tensorcnt.
// ---------------------------------------------------------------------------

#define BSZ   2048
#define TLEN  8192
#define NSEG  32
#define SEGL  (TLEN / NSEG)        // 256
#define ROWS  128                  // rows per block == blockDim.x
#define CHUNK 32                   // timesteps per LDS tile
#define NCH   (SEGL / CHUNK)       // 8 chunks per segment
#define PADS  (CHUNK + 1)          // padded LDS row stride (33 dwords)

typedef unsigned int v4u __attribute__((ext_vector_type(4)));
typedef int          v8i __attribute__((ext_vector_type(8)));
typedef int          v4i __attribute__((ext_vector_type(4)));

// TDM descriptor group1 for a 2D tile: 32 cols x 128 rows of fp32,
// tensor_dim0=TLEN, tensor_dim1=BSZ, dim0 stride=TLEN, padding 1 dw / 32 dw.
__device__ __forceinline__ v8i tdm_group1() {
  v8i g;
  g[0] = (int)((2u << 16) | (1u << 20) | (4u << 22)); // data_size=4B, pad_en, interval=32dw, amount=1dw
  g[1] = (int)(((unsigned)TLEN & 0xFFFFu) << 16);     // tensor_dim0[15:0]
  g[2] = (int)(((unsigned)BSZ & 0xFFFFu) << 16);      // tensor_dim0[31:16]=0, tensor_dim1[15:0]
  g[3] = (int)((unsigned)CHUNK << 16);                // tensor_dim1[31:16]=0, tile_dim0
  g[4] = (int)((unsigned)ROWS);                       // tile_dim1, tile_dim2=0
  g[5] = (int)((unsigned)TLEN);                       // tensor_dim0_stride[31:0]
  g[6] = 0;                                           // stride0 hi, stride1 lo
  g[7] = 0;
  return g;
}

__device__ __forceinline__ void tdm_load_tile(unsigned long long gaddr,
                                              unsigned int lds_off,
                                              v8i g1) {
  v4u g0;
  g0.x = 1u;                                          // count=1, valid user D#
  g0.y = lds_off;                                     // LDS byte offset
  g0.z = (unsigned int)gaddr;                         // global addr [31:0]
  g0.w = ((unsigned int)(gaddr >> 32) & 0x01FFFFFFu)  // global addr [56:32]
         | (2u << 30);                                // type = 2 (image)
  v4i z4 = {0, 0, 0, 0};                              // 2D tile: groups 2/3 unused
  v8i z8 = {0, 0, 0, 0, 0, 0, 0, 0};
  // 6-arg (clang-23 / therock) signature: (g0, g1, g2, g3, g4, cpol)
  __builtin_amdgcn_tensor_load_to_lds(g0, g1, z4, z4, z8, 0);
}

__device__ __forceinline__ unsigned int lds_offset_of(const void* p) {
  // generic LDS pointer = {aperture_hi, lds_offset[31:0]}
  return (unsigned int)(unsigned long long)(uintptr_t)p;
}

// ---------------------------------------------------------------------------
// Pass 1: particular end-state of each segment starting from v = 0.
// ---------------------------------------------------------------------------
__global__ __launch_bounds__(ROWS) void dsvf_pass1(
    const float* __restrict__ x, const float* __restrict__ Gp,
    const float* __restrict__ twoRp, float* __restrict__ vend) {
  const int seg = blockIdx.x;
  const int rowbase = blockIdx.y * ROWS;
  const int tid = threadIdx.x;

  __shared__ float xt[2][ROWS * PADS];

  const float G    = fmaxf(Gp[0], 1e-8f);
  const float twoR = fmaxf(twoRp[0], 0.0f);
  const float c0 = 1.0f / (1.0f + G * (G + twoR));
  const float c1 = G * c0;

  const unsigned long long gbase =
      (unsigned long long)(uintptr_t)x +
      4ull * ((unsigned long long)rowbase * TLEN + (unsigned long long)seg * SEGL);
  const unsigned int lds0 = lds_offset_of(&xt[0][0]);
  const unsigned int lds1 = lds_offset_of(&xt[1][0]);
  const bool issuer = (tid >> 5) == 0;          // wave 0 only (wave32)
  const v8i g1 = tdm_group1();

  if (issuer) tdm_load_tile(gbase, lds0, g1);

  float v0 = 0.0f, v1 = 0.0f;
  for (int c = 0; c < NCH; ++c) {
    if (c + 1 < NCH) {
      if (issuer) {
        tdm_load_tile(gbase + 4ull * (unsigned long long)(c + 1) * CHUNK,
                      ((c + 1) & 1) ? lds1 : lds0, g1);
        __builtin_amdgcn_s_wait_tensorcnt(1);   // chunk c complete (in-order)
      }
    } else {
      if (issuer) __builtin_amdgcn_s_wait_tensorcnt(0);
    }
    __syncthreads();                            // publish LDS tile to all waves

    const float* xs = &xt[c & 1][tid * PADS];
#pragma unroll
    for (int t = 0; t < CHUNK; ++t) {
      const float xv  = xs[t];
      const float imv = xv - v1;
      const float bp  = fmaf(c1, imv, c0 * v0);
      const float lp  = fmaf(G, bp, v1);
      v0 = 2.0f * bp - v0;
      v1 = 2.0f * lp - v1;
    }
    __syncthreads();                            // before buffer is overwritten
  }

  const int r = rowbase + tid;
  vend[(r * NSEG + seg) * 2 + 0] = v0;
  vend[(r * NSEG + seg) * 2 + 1] = v1;
}

// ---------------------------------------------------------------------------
// Pass 2: combine segments.  v_start[s+1] = M^SEGL * v_start[s] + vend_p[s].
// ---------------------------------------------------------------------------
__global__ void dsvf_pass2(const float* __restrict__ vinit,
                           const float* __restrict__ vend,
                           float* __restrict__ vstart,
                           float* __restrict__ vfinal,
                           const float* __restrict__ Gp,
                           const float* __restrict__ twoRp) {
  const int r = blockIdx.x * blockDim.x + threadIdx.x;
  if (r >= BSZ) return;

  const float G    = fmaxf(Gp[0], 1e-8f);
  const float twoR = fmaxf(twoRp[0], 0.0f);
  const float c0 = 1.0f / (1.0f + G * (G + twoR));
  const float c1 = G * c0;

  // M = [[2c0-1, -2c1], [2G c0, 1-2G c1]]  ;  M^SEGL by repeated squaring
  float a = 2.0f * c0 - 1.0f, b = -2.0f * c1;
  float c = 2.0f * G * c0,    d = 1.0f - 2.0f * G * c1;
#pragma unroll
  for (int i = 0; i < 8; ++i) {                 // SEGL = 256 = 2^8
    const float na = a * a + b * c;
    const float nb = a * b + b * d;
    const float nc = c * a + d * c;
    const float nd = c * b + d * d;
    a = na; b = nb; c = nc; d = nd;
  }

  float v0 = vinit[r * 2 + 0];
  float v1 = vinit[r * 2 + 1];
  for (int s = 0; s < NSEG; ++s) {
    vstart[(r * NSEG + s) * 2 + 0] = v0;
    vstart[(r * NSEG + s) * 2 + 1] = v1;
    const float e0 = vend[(r * NSEG + s) * 2 + 0];
    const float e1 = vend[(r * NSEG + s) * 2 + 1];
    const float n0 = a * v0 + b * v1 + e0;
    const float n1 = c * v0 + d * v1 + e1;
    v0 = n0; v1 = n1;
  }
  vfinal[r * 2 + 0] = v0;
  vfinal[r * 2 + 1] = v1;
}

// ---------------------------------------------------------------------------
// Pass 3: rerun each segment from its exact start state, emit y.
// ---------------------------------------------------------------------------
__global__ __launch_bounds__(ROWS) void dsvf_pass3(
    const float* __restrict__ x, const float* __restrict__ vstart,
    float* __restrict__ y,
    const float* __restrict__ Gp, const float* __restrict__ twoRp,
    const float* __restrict__ hpg, const float* __restrict__ bpg,
    const float* __restrict__ lpg, const float* __restrict__ mgp) {
  const int seg = blockIdx.x;
  const int rowbase = blockIdx.y * ROWS;
  const int tid = threadIdx.x;

  __shared__ float xt[2][ROWS * PADS];
  __shared__ float yt[ROWS * PADS];

  const float G    = fmaxf(Gp[0], 1e-8f);
  const float twoR = fmaxf(twoRp[0], 0.0f);
  const float hg   = fminf(fmaxf(hpg[0], -1.0f), 1.0f);
  const float bg   = fmaxf(bpg[0], -1.0f);
  const float lg   = fminf(fmaxf(lpg[0], -1.0f), 1.0f);
  const float mg   = mgp[0];
  const float c0   = 1.0f / (1.0f + G * (G + twoR));
  const float c1   = G * c0;
  const float bg2  = bg * twoR;

  const unsigned long long gbase =
      (unsigned long long)(uintptr_t)x +
      4ull * ((unsigned long long)rowbase * TLEN + (unsigned long long)seg * SEGL);
  const unsigned int lds0 = lds_offset_of(&xt[0][0]);
  const unsigned int lds1 = lds_offset_of(&xt[1][0]);
  const bool issuer = (tid >> 5) == 0;
  const v8i g1 = tdm_group1();

  if (issuer) tdm_load_tile(gbase, lds0, g1);

  const int r = rowbase + tid;
  float v0 = vstart[(r * NSEG + seg) * 2 + 0];
  float v1 = vstart[(r * NSEG + seg) * 2 + 1];

  for (int c = 0; c < NCH; ++c) {
    if (c + 1 < NCH) {
      if (issuer) {
        tdm_load_tile(gbase + 4ull * (unsigned long long)(c + 1) * CHUNK,
                      ((c + 1) & 1) ? lds1 : lds0, g1);
        __builtin_amdgcn_s_wait_tensorcnt(1);
      }
    } else {
      if (issuer) __builtin_amdgcn_s_wait_tensorcnt(0);
    }
    __syncthreads();

    const float* xs = &xt[c & 1][tid * PADS];
    float* ys = &yt[tid * PADS];
#pragma unroll
    for (int t = 0; t < CHUNK; ++t) {
      const float xv  = xs[t];
      const float imv = xv - v1;
      const float bp  = fmaf(c1, imv, c0 * v0);
      const float lp  = fmaf(G, bp, v1);
      const float hp  = xv - lp - twoR * bp;
      ys[t] = mg * (hg * hp + bg2 * bp + lg * lp);
      v0 = 2.0f * bp - v0;
      v1 = 2.0f * lp - v1;
    }
    __syncthreads();                            // y tile complete

    // Coalesced store: 32 lanes cover 32 contiguous floats of one row.
    const size_t colbase = (size_t)seg * SEGL + (size_t)c * CHUNK;
    for (int i = tid; i < ROWS * CHUNK; i += ROWS) {
      const int row = i >> 5;
      const int col = i & (CHUNK - 1);
      y[(size_t)(rowbase + row) * TLEN + colbase + col] = yt[row * PADS + col];
    }
    __syncthreads();                            // before tiles are reused
  }
}

extern "C" void kernel_launch(void* const* d_in, const int* in_sizes, int n_in,
                              void* d_out, int out_size, void* d_ws, size_t ws_size,
                              hipStream_t stream) {
  const float* x    = (const float*)d_in[0];
  const float* vin  = (const float*)d_in[1];
  const float* G    = (const float*)d_in[2];
  const float* twoR = (const float*)d_in[3];
  const float* hpg  = (const float*)d_in[4];
  const float* bpg  = (const float*)d_in[5];
  const float* lpg  = (const float*)d_in[6];
  const float* mg   = (const float*)d_in[7];

  float* y      = (float*)d_out;
  float* vfinal = y + (size_t)BSZ * TLEN;

  float* vend   = (float*)d_ws;                 // [B, NSEG, 2]
  float* vstart = vend + (size_t)BSZ * NSEG * 2;// [B, NSEG, 2]

  dim3 grid(NSEG, BSZ / ROWS);
  dsvf_pass1<<<grid, ROWS, 0, stream>>>(x, G, twoR, vend);
  dsvf_pass2<<<BSZ / 256, 256, 0, stream>>>(vin, vend, vstart, vfinal, G, twoR);
  dsvf_pass3<<<grid, ROWS, 0, stream>>>(x, vstart, y, G, twoR, hpg, bpg, lpg, mg);
}